// AttentionBlock_4148938408372
// MI455X (gfx1250) — compile-verified
//
#include <hip/hip_runtime.h>
#include <hip/hip_bf16.h>

#define B_      16
#define C_      512
#define L_      4096
#define HEADS_  8
#define LH_     512
#define GROUPS_ 32
#define CPG_    16      // channels per group
#define EPS_    1e-5f
#define ATT_SCALE_ 0.125f   // 1/sqrt(C/heads) = 1/sqrt(64)

typedef __attribute__((ext_vector_type(2))) float v2f;
typedef __attribute__((ext_vector_type(8))) float v8f;

// ---------------------------------------------------------------------------
// WMMA 128x128 block-tile core: 8 waves as 4(M) x 2(N); each wave owns a
// 32x64 sub-tile = v8f acc[2][4] (64 VGPRs). BK=16 staged through LDS.
// Per 4-deep K-substep: 2 A-frags + 4 B-frags feed 8 WMMAs
// (1.5 LDS floats per lane per WMMA).
// f32 WMMA 16x16x4 per ISA layout:
//   A frag : VGPR0 = A[M=lane][K+0] (lanes 0-15) / A[M=lane-16][K+2] (16-31)
//   B frag : VGPR0 = B[K+0][N=lane] / B[K+2][N=lane-16]
//   C/D    : VGPR r = D[M=r][N=lane] / D[M=r+8][N=lane-16]
// ---------------------------------------------------------------------------
__device__ __forceinline__ void tile_mma(const float (*As)[17], const float (*Bs)[128],
                                         v8f acc[2][4], int m0, int n0, int lane) {
  const int lm = lane & 15;
  const int kh = (lane >> 4) << 1;   // 0 for lanes 0-15, 2 for lanes 16-31
#pragma unroll
  for (int kk = 0; kk < 16; kk += 4) {
    v2f a[2], b[4];
#pragma unroll
    for (int i = 0; i < 2; i++) {
      a[i].x = As[m0 + i * 16 + lm][kk + kh];
      a[i].y = As[m0 + i * 16 + lm][kk + kh + 1];
    }
#pragma unroll
    for (int j = 0; j < 4; j++) {
      b[j].x = Bs[kk + kh][n0 + j * 16 + lm];
      b[j].y = Bs[kk + kh + 1][n0 + j * 16 + lm];
    }
#pragma unroll
    for (int i = 0; i < 2; i++)
#pragma unroll
      for (int j = 0; j < 4; j++)
        acc[i][j] = __builtin_amdgcn_wmma_f32_16x16x4_f32(false, a[i], false, b[j],
                                                          (short)0, acc[i][j], false, false);
  }
}

// ---------------------------------------------------------------------------
// 1) GroupNorm: one block per (batch, group); 16 ch x 4096 = 65536 elements.
// ---------------------------------------------------------------------------
__global__ void gn_kernel(const float* __restrict__ x, const float* __restrict__ gamma,
                          const float* __restrict__ beta, float* __restrict__ xn) {
  const int bg = blockIdx.x;
  const int b = bg / GROUPS_, g = bg % GROUPS_;
  const size_t base = (size_t)b * C_ * L_ + (size_t)g * CPG_ * L_;
  const int n = CPG_ * L_;
  float s = 0.f, s2 = 0.f;
  for (int i = threadIdx.x; i < n; i += blockDim.x) {
    float v = x[base + i];
    s += v; s2 += v * v;
  }
  __shared__ float red0[256], red1[256];
  red0[threadIdx.x] = s; red1[threadIdx.x] = s2;
  __syncthreads();
  for (int off = 128; off > 0; off >>= 1) {
    if (threadIdx.x < off) {
      red0[threadIdx.x] += red0[threadIdx.x + off];
      red1[threadIdx.x] += red1[threadIdx.x + off];
    }
    __syncthreads();
  }
  const float mu  = red0[0] * (1.0f / n);
  const float var = red1[0] * (1.0f / n) - mu * mu;
  const float inv = rsqrtf(var + EPS_);
  for (int i = threadIdx.x; i < n; i += blockDim.x) {
    const int c = g * CPG_ + (i / L_);
    xn[base + i] = (x[base + i] - mu) * inv * gamma[c] + beta[c];
  }
}

// ---------------------------------------------------------------------------
// 2) QKV GEMM: qkv[b,o,l] = sum_c W[o,c]*xn[b,c,l] + bias[o]
//    grid (L/128, 1536/128, B)
// ---------------------------------------------------------------------------
__global__ void gemm_qkv(const float* __restrict__ W, const float* __restrict__ bias,
                         const float* __restrict__ Xn, float* __restrict__ Out) {
  __shared__ float As[128][17];
  __shared__ float Bs[16][128];
  const int bn = blockIdx.x * 128;
  const int bm = blockIdx.y * 128;
  const int b  = blockIdx.z;
  const float* Bb = Xn + (size_t)b * C_ * L_;
  float* Ob = Out + (size_t)b * 3 * C_ * L_;
  const int t = threadIdx.x, lane = t & 31, wid = t >> 5;
  const int m0 = (wid & 3) * 32, n0 = (wid >> 2) * 64;
  v8f acc[2][4] = {};
  for (int k0 = 0; k0 < C_; k0 += 16) {
    {
      const int r = t >> 1, c0 = (t & 1) * 8;
      const float4 v0 = *(const float4*)(W + (size_t)(bm + r) * C_ + k0 + c0);
      const float4 v1 = *(const float4*)(W + (size_t)(bm + r) * C_ + k0 + c0 + 4);
      As[r][c0]     = v0.x; As[r][c0 + 1] = v0.y; As[r][c0 + 2] = v0.z; As[r][c0 + 3] = v0.w;
      As[r][c0 + 4] = v1.x; As[r][c0 + 5] = v1.y; As[r][c0 + 6] = v1.z; As[r][c0 + 7] = v1.w;
    }
    {
      const int r = t >> 4, c0 = (t & 15) * 8;
      *(float4*)&Bs[r][c0]     = *(const float4*)(Bb + (size_t)(k0 + r) * L_ + bn + c0);
      *(float4*)&Bs[r][c0 + 4] = *(const float4*)(Bb + (size_t)(k0 + r) * L_ + bn + c0 + 4);
    }
    __syncthreads();
    tile_mma(As, Bs, acc, m0, n0, lane);
    __syncthreads();
  }
  const int ln = lane & 15, mh = (lane >> 4) << 3;
#pragma unroll
  for (int i = 0; i < 2; i++)
#pragma unroll
    for (int r = 0; r < 8; r++) {
      const int row = bm + m0 + i * 16 + mh + r;
      const float bv = bias[row];
#pragma unroll
      for (int j = 0; j < 4; j++)
        Ob[(size_t)row * L_ + bn + n0 + j * 16 + ln] = acc[i][j][r] + bv;
    }
}

// ---------------------------------------------------------------------------
// 3) S = scale * Q^T K per (b,h): S[q,k] = scale * sum_c Q[c,q]*K[c,k]
//    Q/K stored in qkv with stride L along c. grid (LH/128, LH/128, B*HEADS)
// ---------------------------------------------------------------------------
__global__ void gemm_qk(const float* __restrict__ QKV, float* __restrict__ S) {
  __shared__ float As[128][17];   // As[q][c]
  __shared__ float Bs[16][128];   // Bs[c][k]
  const int bh = blockIdx.z;
  const int b = bh / HEADS_, h = bh % HEADS_;
  const int bm = blockIdx.y * 128;   // q offset
  const int bn = blockIdx.x * 128;   // k offset
  const float* Qb = QKV + (size_t)b * 3 * C_ * L_ + (size_t)h * LH_;
  const float* Kb = Qb + (size_t)C_ * L_;
  float* Sb = S + (size_t)bh * LH_ * LH_;
  const int t = threadIdx.x, lane = t & 31, wid = t >> 5;
  const int m0 = (wid & 3) * 32, n0 = (wid >> 2) * 64;
  v8f acc[2][4] = {};
  for (int k0 = 0; k0 < C_; k0 += 16) {
    {
      // A column-major in memory: coalesced along q
      const int m = t & 127, c0 = (t >> 7) * 8;
#pragma unroll
      for (int j = 0; j < 8; j++)
        As[m][c0 + j] = Qb[(size_t)(k0 + c0 + j) * L_ + bm + m];
    }
    {
      const int r = t >> 4, c0 = (t & 15) * 8;
      *(float4*)&Bs[r][c0]     = *(const float4*)(Kb + (size_t)(k0 + r) * L_ + bn + c0);
      *(float4*)&Bs[r][c0 + 4] = *(const float4*)(Kb + (size_t)(k0 + r) * L_ + bn + c0 + 4);
    }
    __syncthreads();
    tile_mma(As, Bs, acc, m0, n0, lane);
    __syncthreads();
  }
  const int ln = lane & 15, mh = (lane >> 4) << 3;
#pragma unroll
  for (int i = 0; i < 2; i++)
#pragma unroll
    for (int r = 0; r < 8; r++) {
      const int row = bm + m0 + i * 16 + mh + r;
#pragma unroll
      for (int j = 0; j < 4; j++)
        Sb[(size_t)row * LH_ + bn + n0 + j * 16 + ln] = acc[i][j][r] * ATT_SCALE_;
    }
}

// ---------------------------------------------------------------------------
// 4) Row softmax over 512-long rows; one 256-thread block per row.
// ---------------------------------------------------------------------------
__global__ void softmax_kernel(float* __restrict__ S) {
  float* p = S + (size_t)blockIdx.x * LH_;
  const int t = threadIdx.x;
  float v0 = p[t], v1 = p[t + 256];
  __shared__ float red[256];
  red[t] = fmaxf(v0, v1);
  __syncthreads();
  for (int off = 128; off > 0; off >>= 1) {
    if (t < off) red[t] = fmaxf(red[t], red[t + off]);
    __syncthreads();
  }
  const float m = red[0];
  __syncthreads();
  const float e0 = __expf(v0 - m), e1 = __expf(v1 - m);
  red[t] = e0 + e1;
  __syncthreads();
  for (int off = 128; off > 0; off >>= 1) {
    if (t < off) red[t] += red[t + off];
    __syncthreads();
  }
  const float inv = 1.0f / red[0];
  p[t] = e0 * inv;
  p[t + 256] = e1 * inv;
}

// ---------------------------------------------------------------------------
// 5) A = V * P^T per (b,h): a[c,q] = sum_k P[q,k] * V[c,k]
//    V row-major with lda=L; P(row-major q x k) used column-major as B operand.
//    grid (LH/128, C/128, B*HEADS). Writes into `a` buffer (reuses xn).
// ---------------------------------------------------------------------------
__global__ void gemm_av(const float* __restrict__ QKV, const float* __restrict__ P,
                        float* __restrict__ A) {
  __shared__ float As[128][17];   // As[c][k]
  __shared__ float Bs[16][128];   // Bs[k][q]
  const int bh = blockIdx.z;
  const int b = bh / HEADS_, h = bh % HEADS_;
  const int bm = blockIdx.y * 128;   // c offset
  const int bn = blockIdx.x * 128;   // q offset
  const float* Vb = QKV + (size_t)b * 3 * C_ * L_ + (size_t)2 * C_ * L_ + (size_t)h * LH_;
  const float* Pb = P + (size_t)bh * LH_ * LH_;
  float* Ab = A + (size_t)b * C_ * L_ + (size_t)h * LH_;
  const int t = threadIdx.x, lane = t & 31, wid = t >> 5;
  const int m0 = (wid & 3) * 32, n0 = (wid >> 2) * 64;
  v8f acc[2][4] = {};
  for (int k0 = 0; k0 < LH_; k0 += 16) {
    {
      const int r = t >> 1, c0 = (t & 1) * 8;
      const float4 v0 = *(const float4*)(Vb + (size_t)(bm + r) * L_ + k0 + c0);
      const float4 v1 = *(const float4*)(Vb + (size_t)(bm + r) * L_ + k0 + c0 + 4);
      As[r][c0]     = v0.x; As[r][c0 + 1] = v0.y; As[r][c0 + 2] = v0.z; As[r][c0 + 3] = v0.w;
      As[r][c0 + 4] = v1.x; As[r][c0 + 5] = v1.y; As[r][c0 + 6] = v1.z; As[r][c0 + 7] = v1.w;
    }
    {
      // B[k][n] = P[(bn+n)*LH + k0+k] : column-major source
      const int n = t & 127, c0 = (t >> 7) * 8;
      const float4 v0 = *(const float4*)(Pb + (size_t)(bn + n) * LH_ + k0 + c0);
      const float4 v1 = *(const float4*)(Pb + (size_t)(bn + n) * LH_ + k0 + c0 + 4);
      Bs[c0][n]     = v0.x; Bs[c0 + 1][n] = v0.y; Bs[c0 + 2][n] = v0.z; Bs[c0 + 3][n] = v0.w;
      Bs[c0 + 4][n] = v1.x; Bs[c0 + 5][n] = v1.y; Bs[c0 + 6][n] = v1.z; Bs[c0 + 7][n] = v1.w;
    }
    __syncthreads();
    tile_mma(As, Bs, acc, m0, n0, lane);
    __syncthreads();
  }
  const int ln = lane & 15, mh = (lane >> 4) << 3;
#pragma unroll
  for (int i = 0; i < 2; i++)
#pragma unroll
    for (int r = 0; r < 8; r++) {
      const int row = bm + m0 + i * 16 + mh + r;   // channel c
#pragma unroll
      for (int j = 0; j < 4; j++)
        Ab[(size_t)row * L_ + bn + n0 + j * 16 + ln] = acc[i][j][r];
    }
}

// ---------------------------------------------------------------------------
// 6) Proj GEMM + bias + residual: out[b,o,l] = x + bias[o] + sum_c Wp[o,c]*a[b,c,l]
//    grid (L/128, C/128, B)
// ---------------------------------------------------------------------------
__global__ void gemm_proj(const float* __restrict__ W, const float* __restrict__ bias,
                          const float* __restrict__ A, const float* __restrict__ X,
                          float* __restrict__ Out) {
  __shared__ float As[128][17];
  __shared__ float Bs[16][128];
  const int bn = blockIdx.x * 128;
  const int bm = blockIdx.y * 128;
  const int b  = blockIdx.z;
  const float* Bb = A + (size_t)b * C_ * L_;
  const float* Xb = X + (size_t)b * C_ * L_;
  float* Ob = Out + (size_t)b * C_ * L_;
  const int t = threadIdx.x, lane = t & 31, wid = t >> 5;
  const int m0 = (wid & 3) * 32, n0 = (wid >> 2) * 64;
  v8f acc[2][4] = {};
  for (int k0 = 0; k0 < C_; k0 += 16) {
    {
      const int r = t >> 1, c0 = (t & 1) * 8;
      const float4 v0 = *(const float4*)(W + (size_t)(bm + r) * C_ + k0 + c0);
      const float4 v1 = *(const float4*)(W + (size_t)(bm + r) * C_ + k0 + c0 + 4);
      As[r][c0]     = v0.x; As[r][c0 + 1] = v0.y; As[r][c0 + 2] = v0.z; As[r][c0 + 3] = v0.w;
      As[r][c0 + 4] = v1.x; As[r][c0 + 5] = v1.y; As[r][c0 + 6] = v1.z; As[r][c0 + 7] = v1.w;
    }
    {
      const int r = t >> 4, c0 = (t & 15) * 8;
      *(float4*)&Bs[r][c0]     = *(const float4*)(Bb + (size_t)(k0 + r) * L_ + bn + c0);
      *(float4*)&Bs[r][c0 + 4] = *(const float4*)(Bb + (size_t)(k0 + r) * L_ + bn + c0 + 4);
    }
    __syncthreads();
    tile_mma(As, Bs, acc, m0, n0, lane);
    __syncthreads();
  }
  const int ln = lane & 15, mh = (lane >> 4) << 3;
#pragma unroll
  for (int i = 0; i < 2; i++)
#pragma unroll
    for (int r = 0; r < 8; r++) {
      const int row = bm + m0 + i * 16 + mh + r;
      const float bv = bias[row];
#pragma unroll
      for (int j = 0; j < 4; j++) {
        const size_t idx = (size_t)row * L_ + bn + n0 + j * 16 + ln;
        Ob[idx] = acc[i][j][r] + bv + Xb[idx];
      }
    }
}

// ---------------------------------------------------------------------------
extern "C" void kernel_launch(void* const* d_in, const int* in_sizes, int n_in,
                              void* d_out, int out_size, void* d_ws, size_t ws_size,
                              hipStream_t stream) {
  const float* x        = (const float*)d_in[0];
  const float* gn_gamma = (const float*)d_in[1];
  const float* gn_beta  = (const float*)d_in[2];
  const float* w_qkv    = (const float*)d_in[3];
  const float* b_qkv    = (const float*)d_in[4];
  const float* w_proj   = (const float*)d_in[5];
  const float* b_proj   = (const float*)d_in[6];
  float* out = (float*)d_out;

  // workspace layout (fp32): xn/a (reused) | qkv | att  -> 640 MiB total
  char* ws = (char*)d_ws;
  const size_t xn_bytes  = (size_t)B_ * C_ * L_ * sizeof(float);        // 128 MiB
  const size_t qkv_bytes = 3 * xn_bytes;                                 // 384 MiB
  float* xn  = (float*)ws;                         // also reused as `a`
  float* qkv = (float*)(ws + xn_bytes);
  float* att = (float*)(ws + xn_bytes + qkv_bytes);
  (void)ws_size; (void)in_sizes; (void)n_in; (void)out_size;

  dim3 blk(256);

  // 1) GroupNorm
  gn_kernel<<<dim3(B_ * GROUPS_), blk, 0, stream>>>(x, gn_gamma, gn_beta, xn);

  // 2) QKV projection
  gemm_qkv<<<dim3(L_ / 128, (3 * C_) / 128, B_), blk, 0, stream>>>(w_qkv, b_qkv, xn, qkv);

  // 3) S = scale * Q^T K
  gemm_qk<<<dim3(LH_ / 128, LH_ / 128, B_ * HEADS_), blk, 0, stream>>>(qkv, att);

  // 4) softmax rows
  softmax_kernel<<<dim3(B_ * HEADS_ * LH_), blk, 0, stream>>>(att);

  // 5) A = V * P^T  (writes into xn buffer, which is dead now)
  gemm_av<<<dim3(LH_ / 128, C_ / 128, B_ * HEADS_), blk, 0, stream>>>(qkv, att, xn);

  // 6) proj + bias + residual
  gemm_proj<<<dim3(L_ / 128, C_ / 128, B_), blk, 0, stream>>>(w_proj, b_proj, xn, x, out);
}